// Model_87660282511503
// MI455X (gfx1250) — compile-verified
//
#include <hip/hip_runtime.h>
#include <math.h>

typedef __attribute__((ext_vector_type(2))) float v2f;
typedef __attribute__((ext_vector_type(8))) float v8f;

#define N_NODES 116
#define NPAD    128
#define SEQ     220
#define CHEV    45
#define BS      512

// ---------------------------------------------------------------------------
// WMMA helpers built on V_WMMA_F32_16X16X4_F32.
// A fragment (ISA 7.12.2, 16x4 f32): lanes 0-15 -> M=lane, K={0,1};
// lanes 16-31 -> M=lane-16, K={2,3}.  B mirrors the K split.
// C/D: 8 VGPRs, N = lane&15, M = (lane>>4)*8 + v.
// ---------------------------------------------------------------------------

// single 16x16 tile, B row-major (strided B loads)
__device__ __forceinline__ v8f wmma_gemm_nn(v8f c,
    const float* __restrict__ A, int lda, int rowClamp,
    const float* __restrict__ B, int ldb, int colClamp,
    int K, int row0, int col0)
{
    const int lane = threadIdx.x & 31;
    const int l16  = lane & 15;
    const int koff = (lane >> 4) << 1;          // 0 or 2
    int ar = row0 + l16; if (ar > rowClamp) ar = rowClamp;
    int bc = col0 + l16; if (bc > colClamp) bc = colClamp;
    const float* Ar = A + (size_t)ar * lda;
    for (int k0 = 0; k0 < K; k0 += 4) {
        const int ka = k0 + koff;
        v2f a, b;
        a.x = (ka     < K) ? Ar[ka]     : 0.f;
        a.y = (ka + 1 < K) ? Ar[ka + 1] : 0.f;
        b.x = (ka     < K) ? B[(size_t)ka       * ldb + bc] : 0.f;
        b.y = (ka + 1 < K) ? B[(size_t)(ka + 1) * ldb + bc] : 0.f;
        c = __builtin_amdgcn_wmma_f32_16x16x4_f32(false, a, false, b,
                                                  (short)0, c, false, false);
    }
    return c;
}

// 16x64 strip, B supplied transposed (B[k][n] = Bt[n*ldb+k]); contiguous b64
// loads for both operands, A fragment reused across 4 WMMAs per k-step.
__device__ __forceinline__ void wmma_gemm_nt4(v8f c[4],
    const float* __restrict__ A,  int lda, int rowClamp,
    const float* __restrict__ Bt, int ldb, int colClamp,
    int K, int row0, int col0)
{
    const int lane = threadIdx.x & 31;
    const int l16  = lane & 15;
    const int koff = (lane >> 4) << 1;
    int ar = row0 + l16; if (ar > rowClamp) ar = rowClamp;
    const float* Ar = A + (size_t)ar * lda;
    const float* Bc[4];
#pragma unroll
    for (int j = 0; j < 4; ++j) {
        int bc = col0 + j * 16 + l16; if (bc > colClamp) bc = colClamp;
        Bc[j] = Bt + (size_t)bc * ldb;
    }
    for (int k0 = 0; k0 < K; k0 += 4) {
        const int ka = k0 + koff;
        v2f a;
        a.x = (ka     < K) ? Ar[ka]     : 0.f;
        a.y = (ka + 1 < K) ? Ar[ka + 1] : 0.f;
#pragma unroll
        for (int j = 0; j < 4; ++j) {
            v2f b;
            b.x = (ka     < K) ? Bc[j][ka]     : 0.f;
            b.y = (ka + 1 < K) ? Bc[j][ka + 1] : 0.f;
            c[j] = __builtin_amdgcn_wmma_f32_16x16x4_f32(false, a, false, b,
                                                         (short)0, c[j], false, false);
        }
    }
}

__device__ __forceinline__ void store_tile_guard(float* __restrict__ C, int ldc,
                                                 v8f c, int row0, int col0,
                                                 int M, int N)
{
    const int lane = threadIdx.x & 31;
    const int n  = col0 + (lane & 15);
    const int mb = row0 + ((lane >> 4) << 3);
    if (n < N) {
        for (int v = 0; v < 8; ++v) {
            const int m = mb + v;
            if (m < M) C[(size_t)m * ldc + n] = c[v];
        }
    }
}

// ---------------------------------------------------------------------------
// Stage 1: row-normalize fdata -> Y  (one wave per (b,node) row)
// ---------------------------------------------------------------------------
__global__ void k_norm(const float* __restrict__ fdata, float* __restrict__ Y)
{
    const int row = blockIdx.x;                       // b*116 + n
    const float* x = fdata + (size_t)row * SEQ;
    float*       y = Y     + (size_t)row * SEQ;
    const int t = threadIdx.x;                        // 32 threads
    float s = 0.f, q = 0.f;
    for (int i = t; i < SEQ; i += 32) { float v = x[i]; s += v; q += v * v; }
    for (int o = 16; o > 0; o >>= 1) { s += __shfl_down(s, o, 32); q += __shfl_down(q, o, 32); }
    s = __shfl(s, 0, 32); q = __shfl(q, 0, 32);
    const float mean = s / (float)SEQ;
    const float nrm  = sqrtf(fmaxf(q - s * mean, 0.f));     // ||x - mean||
    const float inv  = 1.f / (nrm + 1e-6f);
    for (int i = t; i < SEQ; i += 32) y[i] = (x[i] - mean) * inv;
}

// ---------------------------------------------------------------------------
// Stage 2: corr = Y * Y^T per batch (stride NPAD); one wave per 16x64 strip
// ---------------------------------------------------------------------------
__global__ void k_corr(const float* __restrict__ Y, float* __restrict__ CORR)
{
    const int tile = blockIdx.x * 4 + (threadIdx.x >> 5);
    if (tile >= BS * 16) return;                      // 8 tm x 2 n-groups
    const int tg = tile & 1;
    const int tm = (tile >> 1) & 7;
    const int b  = tile >> 4;
    const float* Yb = Y + (size_t)b * N_NODES * SEQ;
    v8f c[4] = {};
    wmma_gemm_nt4(c, Yb, SEQ, N_NODES - 1, Yb, SEQ, N_NODES - 1,
                  SEQ, tm * 16, tg * 64);
    float* O = CORR + (size_t)b * NPAD * NPAD;
#pragma unroll
    for (int j = 0; j < 4; ++j)
        store_tile_guard(O, NPAD, c[j], tm * 16, tg * 64 + j * 16,
                         N_NODES, N_NODES);
}

// ---------------------------------------------------------------------------
// Stage 3: P[b,k] = fdata_b @ theta_k   (116x220 @ 220x45)
// ---------------------------------------------------------------------------
__global__ void k_p(const float* __restrict__ fdata, const float* __restrict__ theta,
                    float* __restrict__ P)
{
    const int tile = blockIdx.x * 4 + (threadIdx.x >> 5);
    if (tile >= BS * 3 * 8 * 3) return;
    const int tn = tile % 3;
    int tmp = tile / 3;
    const int tm = tmp & 7; tmp >>= 3;
    const int kk = tmp % 3;
    const int b  = tmp / 3;
    const float* A = fdata + (size_t)b * N_NODES * SEQ;
    const float* B = theta + (size_t)kk * SEQ * CHEV;
    v8f c = {};
    c = wmma_gemm_nn(c, A, SEQ, N_NODES - 1, B, CHEV, CHEV - 1,
                     SEQ, tm * 16, tn * 16);
    store_tile_guard(P + (size_t)(b * 3 + kk) * N_NODES * CHEV, CHEV, c,
                     tm * 16, tn * 16, N_NODES, CHEV);
}

// ---------------------------------------------------------------------------
// Stage 4: corr -> W -> L; power iteration for lam_max; LT = 2L/lam - I,
//          zero-padded to 128x128.  One 128-thread block per batch.
// ---------------------------------------------------------------------------
__global__ void k_lap(const float* __restrict__ CORR, float* __restrict__ LT)
{
    const int b = blockIdx.x;
    const float* C  = CORR + (size_t)b * NPAD * NPAD;
    float*       Lo = LT   + (size_t)b * NPAD * NPAD;
    __shared__ float Ls[N_NODES * N_NODES];
    __shared__ float vsh[128];
    __shared__ float red[128];
    __shared__ float deg[128];
    const int t = threadIdx.x;                        // 128 threads

    if (t < N_NODES) {
        float s = 0.f;
        for (int j = 0; j < N_NODES; ++j) s += (C[t * NPAD + j] + 1.f) * 0.5f;
        deg[t] = s;
    }
    __syncthreads();
    if (t < N_NODES) {
        const float d = deg[t];
        for (int j = 0; j < N_NODES; ++j) {
            const float w = (C[t * NPAD + j] + 1.f) * 0.5f;
            Ls[t * N_NODES + j] = ((t == j) ? d : 0.f) - w;
        }
    }
    // start vector must not be the constant vector (L * 1 == 0)
    vsh[t] = (t < N_NODES) ? (((t & 1) ? 1.f : -1.f) * (1.f + 0.003f * t)) : 0.f;
    __syncthreads();

    float lam = 1.f;
    for (int it = 0; it < 64; ++it) {
        float wv = 0.f;
        if (t < N_NODES) {
            const float* Lr = Ls + t * N_NODES;
            for (int j = 0; j < N_NODES; ++j) wv += Lr[j] * vsh[j];
        }
        red[t] = wv * wv;
        __syncthreads();
        for (int s2 = 64; s2 > 0; s2 >>= 1) {
            if (t < s2) red[t] += red[t + s2];
            __syncthreads();
        }
        lam = sqrtf(red[0] + 1e-30f);
        vsh[t] = (t < N_NODES) ? (wv / lam) : 0.f;
        __syncthreads();
    }

    const float inv = 2.f / lam;
    for (int j = 0; j < NPAD; ++j) {
        float val = 0.f;
        if (t < N_NODES && j < N_NODES)
            val = Ls[t * N_NODES + j] * inv - ((t == j) ? 1.f : 0.f);
        Lo[t * NPAD + j] = val;
    }
}

// ---------------------------------------------------------------------------
// Stage 5: T2 = 2 * LT @ LT - I.  LT is symmetric -> use NT form so both
// operands stream as contiguous b64 loads.  One wave per 16x64 strip.
// ---------------------------------------------------------------------------
__global__ void k_t2(const float* __restrict__ LT, float* __restrict__ T2)
{
    const int tile = blockIdx.x * 4 + (threadIdx.x >> 5);
    if (tile >= BS * 16) return;                      // 8 tm x 2 n-groups
    const int tg = tile & 1;
    const int tm = (tile >> 1) & 7;
    const int b  = tile >> 4;
    const float* L = LT + (size_t)b * NPAD * NPAD;
    v8f c[4] = {};
    wmma_gemm_nt4(c, L, NPAD, NPAD - 1, L, NPAD, NPAD - 1,
                  N_NODES, tm * 16, tg * 64);
    float* O = T2 + (size_t)b * NPAD * NPAD;
    const int lane = threadIdx.x & 31;
    const int mb = tm * 16 + ((lane >> 4) << 3);
#pragma unroll
    for (int j = 0; j < 4; ++j) {
        const int n = tg * 64 + j * 16 + (lane & 15);
        for (int v = 0; v < 8; ++v) {
            const int m = mb + v;
            O[(size_t)m * NPAD + n] = 2.f * c[j][v] - ((m == n) ? 1.f : 0.f);
        }
    }
}

// ---------------------------------------------------------------------------
// Stage 6: H = relu(P0 + LT @ P1 + T2 @ P2)   (116x45 per batch)
// ---------------------------------------------------------------------------
__global__ void k_h(const float* __restrict__ LT, const float* __restrict__ T2,
                    const float* __restrict__ P, float* __restrict__ H)
{
    const int tile = blockIdx.x * 4 + (threadIdx.x >> 5);
    if (tile >= BS * 8 * 3) return;
    const int tn = tile % 3;
    const int tm = (tile / 3) & 7;
    const int b  = tile / 24;
    const float* L  = LT + (size_t)b * NPAD * NPAD;
    const float* T  = T2 + (size_t)b * NPAD * NPAD;
    const float* P0 = P + (size_t)(b * 3 + 0) * N_NODES * CHEV;
    const float* P1 = P + (size_t)(b * 3 + 1) * N_NODES * CHEV;
    const float* P2 = P + (size_t)(b * 3 + 2) * N_NODES * CHEV;
    v8f c = {};
    c = wmma_gemm_nn(c, L, NPAD, NPAD - 1, P1, CHEV, CHEV - 1, N_NODES, tm * 16, tn * 16);
    c = wmma_gemm_nn(c, T, NPAD, NPAD - 1, P2, CHEV, CHEV - 1, N_NODES, tm * 16, tn * 16);
    float* Hb = H + (size_t)b * N_NODES * CHEV;
    const int lane = threadIdx.x & 31;
    const int n  = tn * 16 + (lane & 15);
    const int mb = tm * 16 + ((lane >> 4) << 3);
    if (n < CHEV) {
        for (int v = 0; v < 8; ++v) {
            const int m = mb + v;
            if (m < N_NODES) {
                float val = c[v] + P0[m * CHEV + n];
                Hb[m * CHEV + n] = val > 0.f ? val : 0.f;
            }
        }
    }
}

// ---------------------------------------------------------------------------
// Stage 7: 3x3 VALID conv + bias, flatten -> FLAT (512, 4902)
// ---------------------------------------------------------------------------
__global__ void k_conv(const float* __restrict__ H, const float* __restrict__ cw,
                       const float* __restrict__ cb, float* __restrict__ FLAT)
{
    const int idx = blockIdx.x * blockDim.x + threadIdx.x;
    if (idx >= BS * 114 * 43) return;
    const int cc = idx % 43;
    const int r  = (idx / 43) % 114;
    const int b  = idx / (43 * 114);
    const float* h = H + (size_t)b * N_NODES * CHEV;
    float acc = cb[0];
    for (int i = 0; i < 3; ++i)
        for (int j = 0; j < 3; ++j)
            acc += h[(r + i) * CHEV + (cc + j)] * cw[i * 3 + j];
    FLAT[(size_t)b * 4902 + r * 43 + cc] = acc;
}

// ---------------------------------------------------------------------------
// Stage 8: FC: Z = X(512,K) @ Wt(N,K)^T + bias; one wave per 16x64 strip
// ---------------------------------------------------------------------------
__global__ void k_fc(const float* __restrict__ X, int K,
                     const float* __restrict__ Wt, const float* __restrict__ bias,
                     float* __restrict__ Z, int N, int groupsN)
{
    const int tile = blockIdx.x * 4 + (threadIdx.x >> 5);
    if (tile >= 32 * groupsN) return;
    const int tg = tile % groupsN;
    const int tm = tile / groupsN;
    v8f c[4] = {};
    wmma_gemm_nt4(c, X, K, BS - 1, Wt, K, N - 1, K, tm * 16, tg * 64);
    const int lane = threadIdx.x & 31;
    const int mb = tm * 16 + ((lane >> 4) << 3);
#pragma unroll
    for (int j = 0; j < 4; ++j) {
        const int n = tg * 64 + j * 16 + (lane & 15);
        const float bv = bias[n];
        for (int v = 0; v < 8; ++v) {
            const int m = mb + v;
            Z[(size_t)m * N + n] = c[j][v] + bv;
        }
    }
}

// ---------------------------------------------------------------------------
// Stage 9: batchnorm over the batch axis (one block per feature)
// ---------------------------------------------------------------------------
__global__ void k_bn(const float* __restrict__ X, const float* __restrict__ g,
                     const float* __restrict__ be, float* __restrict__ Y, int Nf)
{
    const int f = blockIdx.x;
    const int t = threadIdx.x;                        // 256 threads
    const float x0 = X[(size_t)t * Nf + f];
    const float x1 = X[(size_t)(t + 256) * Nf + f];
    __shared__ float s1[256], s2[256];
    s1[t] = x0 + x1;
    s2[t] = x0 * x0 + x1 * x1;
    __syncthreads();
    for (int s = 128; s > 0; s >>= 1) {
        if (t < s) { s1[t] += s1[t + s]; s2[t] += s2[t + s]; }
        __syncthreads();
    }
    const float m   = s1[0] * (1.f / 512.f);
    const float var = s2[0] * (1.f / 512.f) - m * m;
    const float sc  = g[f] / sqrtf(var + 1e-5f);
    const float sh  = be[f] - m * sc;
    Y[(size_t)t * Nf + f]         = x0 * sc + sh;
    Y[(size_t)(t + 256) * Nf + f] = x1 * sc + sh;
}

// ---------------------------------------------------------------------------
// Stage 10: logits = Z2n @ w3^T + b3; softmax(axis=1) -> out[0:1024]
// ---------------------------------------------------------------------------
__global__ void k_head(const float* __restrict__ Z2, const float* __restrict__ w3,
                       const float* __restrict__ b3, float* __restrict__ out)
{
    const int r = blockIdx.x * blockDim.x + threadIdx.x;
    if (r >= BS) return;
    const float* z = Z2 + (size_t)r * 256;
    float a0 = b3[0], a1 = b3[1];
    for (int i = 0; i < 256; ++i) {
        const float zi = z[i];
        a0 += zi * w3[i];
        a1 += zi * w3[256 + i];
    }
    const float mx = fmaxf(a0, a1);
    const float e0 = __expf(a0 - mx), e1 = __expf(a1 - mx);
    const float inv = 1.f / (e0 + e1);
    out[r * 2 + 0] = e0 * inv;
    out[r * 2 + 1] = e1 * inv;
}

// ---------------------------------------------------------------------------
// Host launcher
// ---------------------------------------------------------------------------
extern "C" void kernel_launch(void* const* d_in, const int* in_sizes, int n_in,
                              void* d_out, int out_size, void* d_ws, size_t ws_size,
                              hipStream_t stream)
{
    const float* fdata  = (const float*)d_in[0];
    const float* theta  = (const float*)d_in[1];
    const float* conv_w = (const float*)d_in[2];
    const float* conv_b = (const float*)d_in[3];
    const float* w1     = (const float*)d_in[4];
    const float* b1     = (const float*)d_in[5];
    const float* g1     = (const float*)d_in[6];
    const float* be1    = (const float*)d_in[7];
    const float* w2     = (const float*)d_in[8];
    const float* b2     = (const float*)d_in[9];
    const float* g2     = (const float*)d_in[10];
    const float* be2    = (const float*)d_in[11];
    const float* w3     = (const float*)d_in[12];
    const float* b3     = (const float*)d_in[13];
    float* out = (float*)d_out;
    float* ws  = (float*)d_ws;

    // workspace layout (float offsets); CORR region is recycled after k_lap
    const size_t oY    = 0;                                   // 512*116*220  (later P)
    const size_t oC    = (size_t)BS * N_NODES * SEQ;          // 512*128*128
    const size_t oLT   = oC + (size_t)BS * NPAD * NPAD;
    const size_t oT2   = oLT + (size_t)BS * NPAD * NPAD;
    const size_t oP    = oY;                                  // 512*3*116*45 (reuses Y)
    const size_t oH    = oC;                                  // 512*116*45
    const size_t oFLAT = oH   + (size_t)BS * N_NODES * CHEV;  // 512*4902
    const size_t oZ1   = oFLAT + (size_t)BS * 4902;           // 512*1024
    const size_t oZ1n  = oZ1  + (size_t)BS * 1024;
    const size_t oZ2   = oZ1n + (size_t)BS * 1024;            // 512*256

    float* Y    = ws + oY;
    float* CORR = ws + oC;
    float* LT   = ws + oLT;
    float* T2   = ws + oT2;
    float* P    = ws + oP;
    float* H    = ws + oH;
    float* FLAT = ws + oFLAT;
    float* Z1   = ws + oZ1;
    float* Z1n  = ws + oZ1n;
    float* Z2   = ws + oZ2;
    float* Z2n  = out + 1024;                                 // second output, in place

    k_norm<<<BS * N_NODES, 32, 0, stream>>>(fdata, Y);
    k_corr<<<(BS * 16) / 4, 128, 0, stream>>>(Y, CORR);
    k_lap <<<BS, 128, 0, stream>>>(CORR, LT);
    k_p   <<<(BS * 72) / 4, 128, 0, stream>>>(fdata, theta, P);   // Y dead -> P overwrites
    k_t2  <<<(BS * 16) / 4, 128, 0, stream>>>(LT, T2);
    k_h   <<<(BS * 24) / 4, 128, 0, stream>>>(LT, T2, P, H);      // CORR dead -> H region
    k_conv<<<(BS * 114 * 43 + 255) / 256, 256, 0, stream>>>(H, conv_w, conv_b, FLAT);
    k_fc  <<<(32 * 16) / 4, 128, 0, stream>>>(FLAT, 4902, w1, b1, Z1, 1024, 16);
    k_bn  <<<1024, 256, 0, stream>>>(Z1, g1, be1, Z1n, 1024);
    k_fc  <<<(32 * 4) / 4, 128, 0, stream>>>(Z1n, 1024, w2, b2, Z2, 256, 4);
    k_bn  <<<256, 256, 0, stream>>>(Z2, g2, be2, Z2n, 256);
    k_head<<<2, 256, 0, stream>>>(Z2n, w3, b3, out);
}